// ChebConv_62955630625379
// MI455X (gfx1250) — compile-verified
//
#include <hip/hip_runtime.h>

// ---------------- problem constants (from reference) ----------------
#define M_NODES 50000
#define FIN     32
#define NB      4
#define KORD    4
#define CH      32
#define FN      128                 // FIN*NB floats per node
#define ROWS    (NB * M_NODES)      // 200000 GEMM rows
#define TILES_PER_N (M_NODES / 16)  // 3125

typedef __attribute__((ext_vector_type(2))) float v2f;
typedef __attribute__((ext_vector_type(8))) float v8f;

// ---------------- fp32 HW atomic add ----------------
__device__ __forceinline__ void atomAddF32(float* p, float v) {
    unsafeAtomicAdd(p, v);   // lowers to global_atomic_add_f32 on gfx1250
}

// ---------------- 1) x [N,M,Fin] -> x0 [M, Fin*4] (fin-major, batch-minor) ----------------
__global__ void k_transpose(const float* __restrict__ x, float* __restrict__ x0) {
    int idx = blockIdx.x * blockDim.x + threadIdx.x;   // over M*128
    if (idx >= M_NODES * FN) return;
    int m = idx >> 7;
    int c = idx & 127;
    int f = c >> 2;
    int n = c & 3;
    x0[idx] = x[((size_t)n * M_NODES + (size_t)m) * FIN + f];
}

// ---------------- 2) SpMM: dst[row] += val * src[col], 128 floats/node ----------------
// one wave per edge; lane owns float4 -> 512B coalesced gather, 4 f32 atomics
__global__ void k_spmm(const int* __restrict__ rows, const int* __restrict__ cols,
                       const float* __restrict__ vals,
                       const float* __restrict__ src, float* __restrict__ dst, int E) {
    int t = blockIdx.x * blockDim.x + threadIdx.x;
    int e = t >> 5;
    if (e >= E) return;
    int lane = t & 31;
    int r = rows[e];
    int c = cols[e];
    float v = vals[e];
    const float4* s4 = (const float4*)(src + (size_t)c * FN);
    float4 d = s4[lane];
    float* dp = dst + (size_t)r * FN + lane * 4;
    atomAddF32(dp + 0, v * d.x);
    atomAddF32(dp + 1, v * d.y);
    atomAddF32(dp + 2, v * d.z);
    atomAddF32(dp + 3, v * d.w);
}

// ---------------- 3) Chebyshev recurrence: dst = 2*dst - sub ----------------
__global__ void k_combine(float4* __restrict__ dst, const float4* __restrict__ sub) {
    int i = blockIdx.x * blockDim.x + threadIdx.x;     // over M*128/4
    if (i >= M_NODES * FN / 4) return;
    float4 d = dst[i];
    float4 s = sub[i];
    d.x = 2.0f * d.x - s.x;
    d.y = 2.0f * d.y - s.y;
    d.z = 2.0f * d.z - s.z;
    d.w = 2.0f * d.w - s.w;
    dst[i] = d;
}

// ---------------- 4) GEMM: A[200000,128] @ W[128,32] + bias, ReLU ----------------
// A row (n*M+m), col (f*4+kk) = xk[kk][m*128 + f*4 + n]; gathered into LDS.
// Per wave: one 16-row tile, two 16x16 f32 accumulators, 32 x V_WMMA_F32_16X16X4_F32 x2.
#define A_STRIDE 132   // 128+4 pad: A-frag column reads hit 32 distinct banks
#define B_STRIDE 132   // BT row pad keeps v2f loads 8B-aligned and conflict-free

__global__ void __launch_bounds__(128)
k_gemm(const float* __restrict__ xk0, const float* __restrict__ xk1,
       const float* __restrict__ xk2, const float* __restrict__ xk3,
       const float* __restrict__ w, const float* __restrict__ bias,
       float* __restrict__ out) {
    __shared__ float BT[CH * B_STRIDE];        // BT[n][k] = W[k][n]
    __shared__ float A[4][16 * A_STRIDE];      // per-wave 16x128 tile

    const int tid  = threadIdx.x;
    const int wave = tid >> 5;
    const int lane = tid & 31;

    // stage W transposed: coalesced read of w[k*32+n]
    for (int i = tid; i < 128 * CH; i += 128) {
        int k = i >> 5;
        int n = i & 31;
        BT[n * B_STRIDE + k] = w[i];
    }

    const int tile = blockIdx.x * 4 + wave;            // 0..12499
    const int row0 = tile * 16;
    const int nb   = tile / TILES_PER_N;               // batch index (uniform per tile)
    const int m0   = row0 - nb * M_NODES;

    // gather A tile: lane = f; 4 k-order values interleave into consecutive cols
    float* Aw = &A[wave][0];
    for (int i = 0; i < 16; ++i) {
        int base = (m0 + i) * FN + lane * 4 + nb;
        float a0 = xk0[base];
        float a1 = xk1[base];
        float a2 = xk2[base];
        float a3 = xk3[base];
        float* ap = Aw + i * A_STRIDE + lane * 4;
        ap[0] = a0; ap[1] = a1; ap[2] = a2; ap[3] = a3;
    }
    __syncthreads();

    const int mA = lane & 15;     // A frag: m = lane%16
    const int h  = lane >> 4;     // k half select
    v8f c0 = {};
    v8f c1 = {};
    for (int k0 = 0; k0 < 128; k0 += 4) {
        v2f a  = *(const v2f*)(Aw + mA * A_STRIDE + k0 + 2 * h);
        v2f b0 = *(const v2f*)(&BT[mA * B_STRIDE + k0 + 2 * h]);          // channels 0..15
        v2f b1 = *(const v2f*)(&BT[(mA + 16) * B_STRIDE + k0 + 2 * h]);   // channels 16..31
        c0 = __builtin_amdgcn_wmma_f32_16x16x4_f32(false, a, false, b0,
                                                   (short)0, c0, false, false);
        c1 = __builtin_amdgcn_wmma_f32_16x16x4_f32(false, a, false, b1,
                                                   (short)0, c1, false, false);
    }

    // epilogue: bias + ReLU; C/D layout: n = lane%16, m = vgpr + 8*(lane/16)
    const int nOut = lane & 15;
    const float b0v = bias[nOut];
    const float b1v = bias[nOut + 16];
    for (int v = 0; v < 8; ++v) {
        int m = v + 8 * h;
        float r0 = c0[v] + b0v; r0 = r0 > 0.0f ? r0 : 0.0f;
        float r1 = c1[v] + b1v; r1 = r1 > 0.0f ? r1 : 0.0f;
        size_t o = (size_t)(row0 + m) * CH;
        out[o + nOut]      = r0;
        out[o + 16 + nOut] = r1;
    }
}

// ---------------- launcher ----------------
extern "C" void kernel_launch(void* const* d_in, const int* in_sizes, int n_in,
                              void* d_out, int out_size, void* d_ws, size_t ws_size,
                              hipStream_t stream) {
    const float* x      = (const float*)d_in[0];
    const float* L_vals = (const float*)d_in[1];
    const float* w      = (const float*)d_in[2];
    const float* bias   = (const float*)d_in[3];
    const int*   L_rows = (const int*)d_in[4];
    const int*   L_cols = (const int*)d_in[5];
    float* out = (float*)d_out;

    const int E = in_sizes[1];
    const size_t SZ = (size_t)M_NODES * FN;      // floats per Chebyshev buffer
    float* x0 = (float*)d_ws;
    float* x1 = x0 + SZ;
    float* x2 = x1 + SZ;
    float* x3 = x2 + SZ;

    // zero the atomic accumulators (capture-safe)
    hipMemsetAsync(x1, 0, SZ * sizeof(float), stream);
    hipMemsetAsync(x2, 0, SZ * sizeof(float), stream);
    hipMemsetAsync(x3, 0, SZ * sizeof(float), stream);

    // x0 = transpose(x)
    k_transpose<<<(int)((M_NODES * FN + 255) / 256), 256, 0, stream>>>(x, x0);

    const int spmmBlocks = (int)(((long long)E * 32 + 255) / 256);
    const int combBlocks = (int)((M_NODES * FN / 4 + 255) / 256);

    // x1 = L x0
    k_spmm<<<spmmBlocks, 256, 0, stream>>>(L_rows, L_cols, L_vals, x0, x1, E);
    // x2 = 2 L x1 - x0
    k_spmm<<<spmmBlocks, 256, 0, stream>>>(L_rows, L_cols, L_vals, x1, x2, E);
    k_combine<<<combBlocks, 256, 0, stream>>>((float4*)x2, (const float4*)x0);
    // x3 = 2 L x2 - x1
    k_spmm<<<spmmBlocks, 256, 0, stream>>>(L_rows, L_cols, L_vals, x2, x3, E);
    k_combine<<<combBlocks, 256, 0, stream>>>((float4*)x3, (const float4*)x1);

    // out = relu(A @ W + bias) via fp32 WMMA
    k_gemm<<<ROWS / 64, 128, 0, stream>>>(x0, x1, x2, x3, w, bias, out);
}